// NLL_24678882083405
// MI455X (gfx1250) — compile-verified
//
#include <hip/hip_runtime.h>
#include <math.h>

#define NTOT   4096
#define NTHR   1024
#define NPT    4          // elements per thread (NTOT / NTHR)
#define NBINS  1024       // >= NUM_GROUPS (1000), padded; empty bins contribute 0
#define LOG2PI 1.8378770664093453f

typedef __attribute__((ext_vector_type(2))) float v2f;
typedef __attribute__((ext_vector_type(8))) float v8f;

// Full-wave (32 lane) f32 sum using the CDNA5 matrix pipe.
// A(16x4): vgpr0 = per-lane value (K=0 for lanes 0-15, K=2 for lanes 16-31),
// vgpr1 = 0.  B(4x16) = all ones.  => D[m][n] = v[m] + v[m+16] for all n.
// Sum of the 8 D vgprs per lane + xor-16 shuffle gives the full wave sum
// broadcast to every lane.  Must be called with EXEC all ones.
__device__ __forceinline__ float wave_sum32(float v) {
  v2f a; a.x = v;    a.y = 0.0f;
  v2f b; b.x = 1.0f; b.y = 1.0f;
  v8f c = {};
  v8f d = __builtin_amdgcn_wmma_f32_16x16x4_f32(
      /*neg_a=*/false, a, /*neg_b=*/false, b,
      /*c_mod=*/(short)0, c, /*reuse_a=*/false, /*reuse_b=*/false);
  float t = (d[0] + d[1]) + (d[2] + d[3]);
  t += (d[4] + d[5]) + (d[6] + d[7]);
  t += __shfl_xor(t, 16, 32);
  return t;
}

__global__ __launch_bounds__(NTHR) void nll_woodbury_kernel(
    const float* __restrict__ yt, const float* __restrict__ yp,
    const int*   __restrict__ zi, const float* __restrict__ sep,
    const float* __restrict__ sbp, float* __restrict__ out, int n) {

  __shared__ float    s_y[NTOT];       // staged y_true
  __shared__ float    s_p[NTOT];       // staged y_pred
  __shared__ float    s_sum[NBINS];    // per-group residual sums
  __shared__ unsigned s_cnt[NBINS];    // per-group counts
  __shared__ float    s_red[3 * 32];   // per-wave partials (rr, quad, logdet)

  const int tid = threadIdx.x;

  // ---- Phase 0: launch async global->LDS staging (ASYNCcnt path), then
  // overlap bin zeroing + index fetch with the in-flight copies.
  {
    unsigned ldsy = (unsigned)(size_t)(&s_y[tid * NPT]);   // low 32 bits of the
    unsigned ldsp = (unsigned)(size_t)(&s_p[tid * NPT]);   // generic ptr == LDS offset
    unsigned voff = (unsigned)(tid * NPT * 4);             // byte offset, 16B aligned
    unsigned long long by = (unsigned long long)yt;
    unsigned long long bp = (unsigned long long)yp;
    asm volatile("global_load_async_to_lds_b128 %0, %1, %2"
                 :: "v"(ldsy), "v"(voff), "s"(by) : "memory");
    asm volatile("global_load_async_to_lds_b128 %0, %1, %2"
                 :: "v"(ldsp), "v"(voff), "s"(bp) : "memory");
  }

  s_sum[tid] = 0.0f;
  s_cnt[tid] = 0u;
  const int4  z4    = *reinterpret_cast<const int4*>(zi + tid * NPT);
  const float sig2e = sep[0];
  const float sig2b = sbp[0];

  asm volatile("s_wait_asynccnt 0" ::: "memory");
  __syncthreads();

  // ---- Phase 1: residuals + segmented reduction via LDS atomics.
  float rr;
  {
    const int base = tid * NPT;
    const float r0 = s_y[base + 0] - s_p[base + 0];
    const float r1 = s_y[base + 1] - s_p[base + 1];
    const float r2 = s_y[base + 2] - s_p[base + 2];
    const float r3 = s_y[base + 3] - s_p[base + 3];
    rr = r0 * r0 + r1 * r1 + r2 * r2 + r3 * r3;
    atomicAdd(&s_sum[z4.x], r0); atomicAdd(&s_cnt[z4.x], 1u);
    atomicAdd(&s_sum[z4.y], r1); atomicAdd(&s_cnt[z4.y], 1u);
    atomicAdd(&s_sum[z4.z], r2); atomicAdd(&s_cnt[z4.z], 1u);
    atomicAdd(&s_sum[z4.w], r3); atomicAdd(&s_cnt[z4.w], 1u);
  }
  __syncthreads();

  // ---- Phase 2: Woodbury / determinant-lemma per-group terms (1 bin/thread).
  // Empty bins: sg = 0, ng = 0 -> quad = 0, ld = log(1) = 0.
  const float ng   = (float)s_cnt[tid];
  const float sg   = s_sum[tid];
  const float quad = sig2b * sg * sg / fmaf(sig2b, ng, sig2e);
  const float ld   = logf(fmaf(sig2b / sig2e, ng, 1.0f));

  // ---- Phase 3: block reduction: WMMA wave-sum, one LDS hop, WMMA again.
  const float wrr = wave_sum32(rr);
  const float wq  = wave_sum32(quad);
  const float wl  = wave_sum32(ld);
  const int wv = tid >> 5, ln = tid & 31;
  if (ln == 0) { s_red[wv] = wrr; s_red[32 + wv] = wq; s_red[64 + wv] = wl; }
  __syncthreads();
  // All 32 waves redundantly reduce the 32 per-wave partials (keeps EXEC full,
  // no divergence around the WMMAs).
  const float RR = wave_sum32(s_red[ln]);
  const float QD = wave_sum32(s_red[32 + ln]);
  const float LD = wave_sum32(s_red[64 + ln]);

  if (tid == 0) {
    const float fn     = (float)n;
    const float logdet = fn * logf(sig2e) + LD;      // det lemma
    const float loss2  = (RR - QD) / sig2e;          // Woodbury quadratic form
    out[0] = 0.5f * (fn * LOG2PI + logdet + loss2);
  }
}

extern "C" void kernel_launch(void* const* d_in, const int* in_sizes, int n_in,
                              void* d_out, int out_size, void* d_ws, size_t ws_size,
                              hipStream_t stream) {
  const float* yt = (const float*)d_in[0];
  const float* yp = (const float*)d_in[1];
  const int*   zi = (const int*)d_in[2];
  const float* se = (const float*)d_in[3];
  const float* sb = (const float*)d_in[4];
  float* out = (float*)d_out;
  const int n = in_sizes[0];   // 4096
  nll_woodbury_kernel<<<1, NTHR, 0, stream>>>(yt, yp, zi, se, sb, out, n);
}